// DeeperGCN_24696061952411
// MI455X (gfx1250) — compile-verified
//
#include <hip/hip_runtime.h>
#include <hip/hip_bf16.h>

#define HCH   64
#define HCH2  128
#define GEPS  1e-7f
#define LNEPS 1e-5f

typedef float v2f __attribute__((ext_vector_type(2)));
typedef float v8f __attribute__((ext_vector_type(8)));

// ---- monotone float<->uint mapping so unsigned atomicMax == float max ----
__device__ __forceinline__ unsigned fmap(float x) {
    unsigned b = __float_as_uint(x);
    return (b & 0x80000000u) ? ~b : (b | 0x80000000u);
}
__device__ __forceinline__ float funmap(unsigned u) {
    unsigned b = (u & 0x80000000u) ? (u & 0x7fffffffu) : ~u;
    return __uint_as_float(b);
}
#define NEG_INF_MAPPED 0x007FFFFFu   // fmap(-inf)

// ---------------- elementwise / edge kernels ----------------
__global__ void init_kernel(unsigned* __restrict__ mx, float* __restrict__ num,
                            float* __restrict__ den, int n) {
    int i = blockIdx.x * blockDim.x + threadIdx.x;
    if (i < n) { mx[i] = NEG_INF_MAPPED; num[i] = 0.f; den[i] = 0.f; }
}

__global__ void edge_max_kernel(const float* __restrict__ y, const int* __restrict__ ei,
                                const float* __restrict__ tptr, unsigned* __restrict__ mx,
                                int E, int total) {
    int i = blockIdx.x * blockDim.x + threadIdx.x;
    if (i >= total) return;
    int e = i >> 6, c = i & 63;
    float tv = tptr[0];
    int src = ei[e], dst = ei[E + e];
    float m = fmaxf(y[src * HCH + c], 0.f) + GEPS;
    atomicMax(&mx[dst * HCH + c], fmap(tv * m));
}

__global__ void fix_max_kernel(float* __restrict__ mx, int n) {
    int i = blockIdx.x * blockDim.x + threadIdx.x;
    if (i >= n) return;
    unsigned u = ((unsigned*)mx)[i];
    float f = funmap(u);
    if ((__float_as_uint(f) & 0x7f800000u) == 0x7f800000u) f = 0.f;  // inf/NaN -> 0
    mx[i] = f;
}

__global__ void edge_sum_kernel(const float* __restrict__ y, const int* __restrict__ ei,
                                const float* __restrict__ tptr, const float* __restrict__ zmax,
                                float* __restrict__ num, float* __restrict__ den,
                                int E, int total) {
    int i = blockIdx.x * blockDim.x + threadIdx.x;
    if (i >= total) return;
    int e = i >> 6, c = i & 63;
    float tv = tptr[0];
    int src = ei[e], dst = ei[E + e];
    float m = fmaxf(y[src * HCH + c], 0.f) + GEPS;
    int nc = dst * HCH + c;
    float a = __expf(tv * m - zmax[nc]);
    atomicAdd(&den[nc], a);
    atomicAdd(&num[nc], m * a);
}

__global__ void node_out_kernel(const float* __restrict__ num, const float* __restrict__ den,
                                const float* __restrict__ y, float* __restrict__ out, int n) {
    int i = blockIdx.x * blockDim.x + threadIdx.x;
    if (i < n) out[i] = num[i] / fmaxf(den[i], GEPS) + y[i];
}

// row LayerNorm(H=64) + ReLU, one thread per node
__global__ void ln_relu_kernel(const float* __restrict__ h, const float* __restrict__ g,
                               const float* __restrict__ b, float* __restrict__ y, int N) {
    int n = blockIdx.x * blockDim.x + threadIdx.x;
    if (n >= N) return;
    const float* r = h + (long)n * HCH;
    float s = 0.f, q = 0.f;
    for (int c = 0; c < HCH; ++c) { float v = r[c]; s += v; q += v * v; }
    float mu = s * (1.f / HCH);
    float var = q * (1.f / HCH) - mu * mu;
    float rs = rsqrtf(var + LNEPS);
    float* o = y + (long)n * HCH;
    for (int c = 0; c < HCH; ++c) {
        float v = (r[c] - mu) * rs * g[c] + b[c];
        o[c] = fmaxf(v, 0.f);
    }
}

// ---------------- WMMA GEMM kernels (fp32, 16x16x4) ----------------
// generic: C[N,64] = A[N,K] @ W[K,64] + bias, optional accumulate (residual)
__global__ __launch_bounds__(128)
void gemm64_kernel(const float* __restrict__ A, const float* __restrict__ W,
                   const float* __restrict__ bias, float* __restrict__ C,
                   int K, int accum) {
    int lane = threadIdx.x & 31;
    int wave = threadIdx.x >> 5;          // 0..3 -> 16-col slice
    int m0 = blockIdx.x * 16;
    int ra = lane & 15;
    int ks = (lane >> 4) << 1;            // 0 or 2
    int col = wave * 16 + (lane & 15);
    v8f acc = {0.f,0.f,0.f,0.f,0.f,0.f,0.f,0.f};
    for (int k0 = 0; k0 < K; k0 += 4) {
        v2f a, b;
        const float* ap = A + (long)(m0 + ra) * K + k0 + ks;
        a.x = ap[0]; a.y = ap[1];
        b.x = W[(k0 + ks) * 64 + col];
        b.y = W[(k0 + ks + 1) * 64 + col];
        acc = __builtin_amdgcn_wmma_f32_16x16x4_f32(false, a, false, b,
                                                    (short)0, acc, false, false);
    }
    float bv = bias[col];
#pragma unroll
    for (int r = 0; r < 8; ++r) {
        int row = m0 + ((lane < 16) ? r : r + 8);
        long idx = (long)row * 64 + col;
        float v = acc[r] + bv;
        C[idx] = accum ? (C[idx] + v) : v;
    }
}

// fused: U = relu(LayerNorm(A[N,64] @ W[64,128] + bias))  (128-wide LN via LDS tile)
__global__ __launch_bounds__(256)
void gemm_ln_relu_kernel(const float* __restrict__ A, const float* __restrict__ W,
                         const float* __restrict__ bias, const float* __restrict__ g,
                         const float* __restrict__ bvec, float* __restrict__ U) {
    __shared__ float tile[16][HCH2 + 1];
    __shared__ float smu[16], srs[16];
    int lane = threadIdx.x & 31;
    int wave = threadIdx.x >> 5;          // 0..7 -> 16-col slice of 128
    int m0 = blockIdx.x * 16;
    int ra = lane & 15;
    int ks = (lane >> 4) << 1;
    int col = wave * 16 + (lane & 15);
    v8f acc = {0.f,0.f,0.f,0.f,0.f,0.f,0.f,0.f};
    for (int k0 = 0; k0 < 64; k0 += 4) {
        v2f a, b;
        const float* ap = A + (long)(m0 + ra) * 64 + k0 + ks;
        a.x = ap[0]; a.y = ap[1];
        b.x = W[(k0 + ks) * HCH2 + col];
        b.y = W[(k0 + ks + 1) * HCH2 + col];
        acc = __builtin_amdgcn_wmma_f32_16x16x4_f32(false, a, false, b,
                                                    (short)0, acc, false, false);
    }
    float bv = bias[col];
#pragma unroll
    for (int r = 0; r < 8; ++r) {
        int row = (lane < 16) ? r : r + 8;
        tile[row][col] = acc[r] + bv;
    }
    __syncthreads();
    if (threadIdx.x < 16) {
        float s = 0.f, q = 0.f;
        for (int c = 0; c < HCH2; ++c) { float v = tile[threadIdx.x][c]; s += v; q += v * v; }
        float mu = s * (1.f / HCH2);
        float var = q * (1.f / HCH2) - mu * mu;
        smu[threadIdx.x] = mu;
        srs[threadIdx.x] = rsqrtf(var + LNEPS);
    }
    __syncthreads();
    for (int idx = threadIdx.x; idx < 16 * HCH2; idx += 256) {
        int row = idx >> 7, c = idx & 127;
        float v = (tile[row][c] - smu[row]) * srs[row] * g[c] + bvec[c];
        U[(long)(m0 + row) * HCH2 + c] = fmaxf(v, 0.f);
    }
}

// n_out = A[N,64] @ nW[64,17] + nb  (pad cols to 32, mask loads/stores)
__global__ __launch_bounds__(64)
void nout_kernel(const float* __restrict__ A, const float* __restrict__ W,
                 const float* __restrict__ bias, float* __restrict__ out) {
    int lane = threadIdx.x & 31;
    int wave = threadIdx.x >> 5;          // 0..1
    int m0 = blockIdx.x * 16;
    int ra = lane & 15;
    int ks = (lane >> 4) << 1;
    int col = wave * 16 + (lane & 15);    // 0..31, valid < 17
    v8f acc = {0.f,0.f,0.f,0.f,0.f,0.f,0.f,0.f};
    for (int k0 = 0; k0 < 64; k0 += 4) {
        v2f a, b;
        const float* ap = A + (long)(m0 + ra) * 64 + k0 + ks;
        a.x = ap[0]; a.y = ap[1];
        b.x = (col < 17) ? W[(k0 + ks) * 17 + col] : 0.f;
        b.y = (col < 17) ? W[(k0 + ks + 1) * 17 + col] : 0.f;
        acc = __builtin_amdgcn_wmma_f32_16x16x4_f32(false, a, false, b,
                                                    (short)0, acc, false, false);
    }
    if (col < 17) {
        float bv = bias[col];
#pragma unroll
        for (int r = 0; r < 8; ++r) {
            int row = m0 + ((lane < 16) ? r : r + 8);
            out[(long)row * 17 + col] = acc[r] + bv;
        }
    }
}

// per-tile column max of A[N,64] @ gW[64,16] + gb  -> partial[blk*16 + c]
__global__ __launch_bounds__(32)
void gmax_kernel(const float* __restrict__ A, const float* __restrict__ W,
                 const float* __restrict__ bias, float* __restrict__ partial) {
    int lane = threadIdx.x;
    int m0 = blockIdx.x * 16;
    int ra = lane & 15;
    int ks = (lane >> 4) << 1;
    int col = lane & 15;
    v8f acc = {0.f,0.f,0.f,0.f,0.f,0.f,0.f,0.f};
    for (int k0 = 0; k0 < 64; k0 += 4) {
        v2f a, b;
        const float* ap = A + (long)(m0 + ra) * 64 + k0 + ks;
        a.x = ap[0]; a.y = ap[1];
        b.x = W[(k0 + ks) * 16 + col];
        b.y = W[(k0 + ks + 1) * 16 + col];
        acc = __builtin_amdgcn_wmma_f32_16x16x4_f32(false, a, false, b,
                                                    (short)0, acc, false, false);
    }
    float bv = bias[col];
    float mx = -INFINITY;
#pragma unroll
    for (int r = 0; r < 8; ++r) mx = fmaxf(mx, acc[r] + bv);
    mx = fmaxf(mx, __shfl_xor(mx, 16, 32));   // combine rows 0-7 with 8-15 (same col)
    if (lane < 16) partial[blockIdx.x * 16 + lane] = mx;
}

__global__ void greduce_kernel(const float* __restrict__ partial, float* __restrict__ out, int nb) {
    int c = threadIdx.x;
    if (c >= 16) return;
    float m = -INFINITY;
    for (int b = 0; b < nb; ++b) m = fmaxf(m, partial[b * 16 + c]);
    out[c] = m;
}

// ---------------- launch ----------------
extern "C" void kernel_launch(void* const* d_in, const int* in_sizes, int n_in,
                              void* d_out, int out_size, void* d_ws, size_t ws_size,
                              hipStream_t stream) {
    (void)n_in; (void)out_size; (void)ws_size;
    const float* x      = (const float*)d_in[0];
    const float* node_W = (const float*)d_in[1];
    const float* node_b = (const float*)d_in[2];
    const float* W1     = (const float*)d_in[3];   // [L,64,128]
    const float* b1     = (const float*)d_in[4];   // [L,128]
    const float* mg     = (const float*)d_in[5];   // [L,128]
    const float* mb     = (const float*)d_in[6];
    const float* W2     = (const float*)d_in[7];   // [L,128,64]
    const float* b2     = (const float*)d_in[8];   // [L,64]
    const float* t      = (const float*)d_in[9];   // [L]
    const float* lng    = (const float*)d_in[10];  // [L,64]
    const float* lnb    = (const float*)d_in[11];
    const int*   ei     = (const int*)d_in[12];    // [2,E]
    const float* gW     = (const float*)d_in[13];  // [64,16]
    const float* gb     = (const float*)d_in[14];
    const float* nW     = (const float*)d_in[15];  // [64,17]
    const float* nb     = (const float*)d_in[16];

    int N = in_sizes[0] / 32;
    int E = in_sizes[12] / 2;

    float* ws = (float*)d_ws;
    float*    bufH    = ws;                       // [N,64]
    float*    bufY    = bufH + (long)N * 64;      // [N,64]
    float*    bufMax  = bufY + (long)N * 64;      // [N,64] (uint during max phase; also "out")
    unsigned* bufMaxU = (unsigned*)bufMax;
    float*    bufNum  = bufMax + (long)N * 64;    // [N,64]
    float*    bufDen  = bufNum + (long)N * 64;    // [N,64]
    float*    bufU    = bufDen + (long)N * 64;    // [N,128]
    float*    partial = bufU + (long)N * 128;     // [N/16,16]

    int nTiles = N / 16;
    int nodeTot = N * 64;
    int edgeTot = E * 64;
    int nodeBlk = (nodeTot + 255) / 256;
    int edgeBlk = (edgeTot + 255) / 256;

    // node encoder: h = x @ node_W + node_b   (K=32)
    gemm64_kernel<<<nTiles, 128, 0, stream>>>(x, node_W, node_b, bufH, 32, 0);

    for (int i = 0; i < 4; ++i) {
        const float* yin = (i == 0) ? bufH : bufY;
        if (i > 0)
            ln_relu_kernel<<<(N + 255) / 256, 256, 0, stream>>>(bufH, lng + i * 64, lnb + i * 64, bufY, N);

        init_kernel<<<nodeBlk, 256, 0, stream>>>(bufMaxU, bufNum, bufDen, nodeTot);
        edge_max_kernel<<<edgeBlk, 256, 0, stream>>>(yin, ei, t + i, bufMaxU, E, edgeTot);
        fix_max_kernel<<<nodeBlk, 256, 0, stream>>>(bufMax, nodeTot);
        edge_sum_kernel<<<edgeBlk, 256, 0, stream>>>(yin, ei, t + i, bufMax, bufNum, bufDen, E, edgeTot);
        node_out_kernel<<<nodeBlk, 256, 0, stream>>>(bufNum, bufDen, yin, bufMax, nodeTot);

        // MLP: u = relu(LN(out @ W1 + b1)); h (+)= u @ W2 + b2
        gemm_ln_relu_kernel<<<nTiles, 256, 0, stream>>>(bufMax, W1 + i * 64 * 128, b1 + i * 128,
                                                        mg + i * 128, mb + i * 128, bufU);
        gemm64_kernel<<<nTiles, 128, 0, stream>>>(bufU, W2 + i * 128 * 64, b2 + i * 64, bufH,
                                                  128, (i > 0) ? 1 : 0);
    }

    // final norm + heads
    ln_relu_kernel<<<(N + 255) / 256, 256, 0, stream>>>(bufH, lng, lnb, bufY, N);
    nout_kernel<<<nTiles, 64, 0, stream>>>(bufY, nW, nb, (float*)d_out + 16);
    gmax_kernel<<<nTiles, 32, 0, stream>>>(bufY, gW, gb, partial);
    greduce_kernel<<<1, 32, 0, stream>>>(partial, (float*)d_out, nTiles);
}